// Attn_47605417509047
// MI455X (gfx1250) — compile-verified
//
#include <hip/hip_runtime.h>

typedef float v2f __attribute__((ext_vector_type(2)));
typedef float v4f __attribute__((ext_vector_type(4)));
typedef float v8f __attribute__((ext_vector_type(8)));

#define LMAX 2048
#define BATCH 64
#define HID 1024

// ---------------------------------------------------------------------------
// Kernel 1: ha[b, 0..2] = h[b,:] @ affect_matrix[:, 0..2]   via f32 WMMA
// One block, 4 waves; wave w handles rows b = 16w .. 16w+15.
// A (16x4 f32): lane m = lane&15 holds K = k0 + 2*(lane>>4) + {0,1} in vgpr {0,1}
// B (4x16 f32): lane n = lane&15 holds K = k0 + 2*(lane>>4) + {0,1} in vgpr {0,1}
// C/D (16x16 f32): vgpr r -> M = r + 8*(lane>>4), N = lane&15
// B columns >= 3 are zeroed with a loop-invariant multiplicative mask so the
// inner loop has no exec-mask branching around the loads.
// ---------------------------------------------------------------------------
__global__ __launch_bounds__(128) void affect_gemm_wmma(
    const float* __restrict__ h, const float* __restrict__ aff,
    float* __restrict__ ha /* [BATCH*4] */) {
  const int wave = threadIdx.x >> 5;
  const int lane = threadIdx.x & 31;
  const int b0   = wave * 16;
  const int mn   = lane & 15;         // M for A rows, N for B cols
  const int kp   = (lane >> 4) << 1;  // 0 or 2
  const bool  ncol = (mn < 3);
  const int   nc   = ncol ? mn : 0;   // clamped column -> loads always in-bounds
  const float msk  = ncol ? 1.0f : 0.0f;

  const float* hrow = h + (size_t)(b0 + mn) * HID + kp;

  v8f c = {};
  for (int k0 = 0; k0 < HID; k0 += 4) {
    v2f a;
    a.x = hrow[k0];
    a.y = hrow[k0 + 1];
    v2f bm;
    bm.x = aff[(size_t)(k0 + kp    ) * 3 + nc] * msk;
    bm.y = aff[(size_t)(k0 + kp + 1) * 3 + nc] * msk;
    c = __builtin_amdgcn_wmma_f32_16x16x4_f32(
        /*neg_a=*/false, a, /*neg_b=*/false, bm,
        /*c_mod=*/(short)0, c, /*reuse_a=*/false, /*reuse_b=*/false);
  }

  if (ncol) {
    const int mbase = b0 + ((lane >> 4) << 3);
#pragma unroll
    for (int r = 0; r < 8; ++r)
      ha[(size_t)(mbase + r) * 4 + mn] = c[r];
  }
}

// ---------------------------------------------------------------------------
// Kernel 2: energies[b, l] = h[b]·enc[l,b] + sum_n ha[b,n]*emb[l,b,n]
// Bandwidth-bound: streams 537MB of enc once. Block = (8 l's, 1 b), 8 waves.
// hidden[b] staged in LDS (reused by all 8 waves); enc loads non-temporal.
// ---------------------------------------------------------------------------
__global__ __launch_bounds__(256) void dot_energy(
    const float* __restrict__ hidden, const float* __restrict__ enc,
    const float* __restrict__ emb, const float* __restrict__ ha,
    float* __restrict__ energies /* [BATCH, LMAX] */) {
  __shared__ float sh[HID];
  const int b   = blockIdx.y;
  const int tid = threadIdx.x;

  // cooperative load of hidden[b] (1024 floats, 256 threads x float4)
  {
    const v4f* hv = (const v4f*)(hidden + (size_t)b * HID);
    ((v4f*)sh)[tid] = hv[tid];
  }
  __syncthreads();

  const int wave = tid >> 5;
  const int lane = tid & 31;
  const int l    = blockIdx.x * 8 + wave;

  const v4f* ev = (const v4f*)(enc + ((size_t)l * BATCH + b) * HID);
  const v4f* hv = (const v4f*)sh;

  float p = 0.0f;
#pragma unroll
  for (int c = 0; c < 8; ++c) {
    v4f e  = __builtin_nontemporal_load(ev + c * 32 + lane);
    v4f hh = hv[c * 32 + lane];
    p = fmaf(e.x, hh.x, p);
    p = fmaf(e.y, hh.y, p);
    p = fmaf(e.z, hh.z, p);
    p = fmaf(e.w, hh.w, p);
  }

  // fold affect score into lanes 0..2 before the reduction
  if (lane < 3)
    p += ha[b * 4 + lane] * emb[((size_t)l * BATCH + b) * 3 + lane];

#pragma unroll
  for (int off = 16; off; off >>= 1)
    p += __shfl_xor(p, off, 32);

  if (lane == 0)
    energies[(size_t)b * LMAX + l] = p;
}

// ---------------------------------------------------------------------------
// Kernel 3: in-place softmax over L per row b. One block per b.
// ---------------------------------------------------------------------------
__global__ __launch_bounds__(256) void softmax_rows(float* __restrict__ io) {
  __shared__ float red[256];
  const int b = blockIdx.x;
  const int t = threadIdx.x;
  float* row = io + (size_t)b * LMAX;

  float e[8];
  float m = -INFINITY;
#pragma unroll
  for (int i = 0; i < 8; ++i) {
    e[i] = row[t + i * 256];
    m = fmaxf(m, e[i]);
  }
  red[t] = m;
  __syncthreads();
  for (int s = 128; s; s >>= 1) {
    if (t < s) red[t] = fmaxf(red[t], red[t + s]);
    __syncthreads();
  }
  m = red[0];
  __syncthreads();

  float sum = 0.0f;
#pragma unroll
  for (int i = 0; i < 8; ++i) {
    e[i] = expf(e[i] - m);
    sum += e[i];
  }
  red[t] = sum;
  __syncthreads();
  for (int s = 128; s; s >>= 1) {
    if (t < s) red[t] += red[t + s];
    __syncthreads();
  }
  const float inv = 1.0f / red[0];

#pragma unroll
  for (int i = 0; i < 8; ++i)
    row[t + i * 256] = e[i] * inv;
}

// ---------------------------------------------------------------------------
extern "C" void kernel_launch(void* const* d_in, const int* in_sizes, int n_in,
                              void* d_out, int out_size, void* d_ws, size_t ws_size,
                              hipStream_t stream) {
  const float* hidden = (const float*)d_in[0];  // [1, B, H]
  const float* enc    = (const float*)d_in[1];  // [L, B, H]
  const float* emb    = (const float*)d_in[2];  // [L, B, 3]
  const float* aff    = (const float*)d_in[3];  // [H, 3]
  float* out = (float*)d_out;                   // [B, 1, L] == B*L floats
  float* ha  = (float*)d_ws;                    // [B*4] floats (1 KB)

  affect_gemm_wmma<<<1, 128, 0, stream>>>(hidden, aff, ha);
  dot_energy<<<dim3(LMAX / 8, BATCH), 256, 0, stream>>>(hidden, enc, emb, ha, out);
  softmax_rows<<<BATCH, 256, 0, stream>>>(out);
}